// YOLO3Loss_35708358099423
// MI455X (gfx1250) — compile-verified
//
#include <hip/hip_runtime.h>
#include <hip/hip_bf16.h>
#include <math.h>

#define N_TGT 50
#define NCLS  80
#define EPSF  1e-12f

typedef float v2f __attribute__((ext_vector_type(2)));
typedef float v8f __attribute__((ext_vector_type(8)));

struct __align__(16) Rec {
    int   idx;    // scatter index in [0,A), or -1 if invalid/dropped
    int   cls;
    float tx, ty, tw, th, wwh, tconf;
};

// SCALED_ANCHORS = ANCHORS / [32,16,8]  (all exactly representable)
__constant__ float SA_W[3][3] = {{3.625f, 4.875f, 11.65625f},
                                 {1.875f, 3.875f, 3.6875f},
                                 {1.25f,  2.0f,   4.125f}};
__constant__ float SA_H[3][3] = {{2.8125f, 6.1875f, 10.1875f},
                                 {3.8125f, 2.8125f, 7.4375f},
                                 {1.625f,  3.75f,   2.875f}};

__device__ __forceinline__ float clip_p(float v) {
    // 1.0f - 1e-12f folds to 1.0f: matches jnp.clip(p, EPS, 1-EPS) in fp32
    return fminf(fmaxf(v, EPSF), 1.0f - EPSF);
}

// Wave-wide (32-lane) sum using V_WMMA_F32_16X16X4_F32.
// A = per-lane partials (2 f32/lane = full 16x4 A matrix), B = ones(4x16).
// D[i][j] = sum_k A[i][k]  ->  total = sum_i D[i][0].
// Column N=0 of C/D lives in lanes 0 (M=0..7) and 16 (M=8..15), VGPRs 0..7.
// Must be called from convergent code (EXEC all ones).
__device__ __forceinline__ float wave_sum_wmma(float s) {
    v2f a;  a.x = s;    a.y = 0.0f;
    v2f bo; bo.x = 1.0f; bo.y = 1.0f;     // all-ones B: layout agnostic
    v8f c = {};
    c = __builtin_amdgcn_wmma_f32_16x16x4_f32(
            false, a, false, bo, (short)0, c, false, false);
    float r = c[0]+c[1]+c[2]+c[3]+c[4]+c[5]+c[6]+c[7];
    float t0  = __shfl(r, 0, 32);
    float t16 = __shfl(r, 16, 32);
    return t0 + t16;
}

// ---------------- kernel 0: init hit flags + accumulators ----------------
__global__ void k_init(unsigned int* hit_words, long n_words, float* acc) {
    long i  = (long)blockIdx.x * blockDim.x + threadIdx.x;
    long gs = (long)gridDim.x * blockDim.x;
    for (; i < n_words; i += gs) hit_words[i] = 0u;
    if (blockIdx.x == 0 && threadIdx.x < 8) acc[threadIdx.x] = 0.0f;
}

// ---------------- kernel 1: per-target records + ignore("hit") flags -----
__global__ void k_targets(const float* __restrict__ tgt,
                          const int*   __restrict__ stride_p,
                          Rec* __restrict__ recs,
                          unsigned char* __restrict__ hit,
                          int B, int A) {
    int b = blockIdx.x;
    int n = threadIdx.x;
    if (n >= N_TGT) return;

    int   stride = stride_p[0];
    float base   = 416.0f / (float)stride;
    int   fs[3], ln[3];
    for (int i = 0; i < 3; ++i) fs[i] = (int)(base * (float)(1 << i));
    ln[0] = 0;
    ln[1] = 3 * fs[0] * fs[0];
    ln[2] = ln[1] + 3 * fs[1] * fs[1];

    const float* t = tgt + ((long)b * N_TGT + n) * 5;
    float t0 = t[0], t1 = t[1], t2 = t[2], t3 = t[3], t4 = t[4];
    bool valid = (t0 + t1 + t2 + t3 + t4) != 0.0f;

    float bestIouM[3]; int bestAM[3], gim[3], gjm[3];
    for (int m = 0; m < 3; ++m) {
        float f  = (float)fs[m];
        float gw = t3 * f, gh = t4 * f;
        int gi = (int)floorf(t1 * f);
        int gj = (int)floorf(t2 * f);
        gim[m] = gi; gjm[m] = gj;
        float bi = -1.0f; int ba = 0;
        for (int k = 0; k < 3; ++k) {
            float aw = SA_W[m][k], ah = SA_H[m][k];
            float inter = fminf(gw, aw) * fminf(gh, ah);
            float uni   = gw * gh + aw * ah - inter;
            float iou   = inter / (uni + 1e-16f);
            if (iou > bi) { bi = iou; ba = k; }        // first-max = jnp.argmax
            if (valid && iou > 0.5f) {                 // ignore-region scatter
                int idxno = ln[m] + 3 * gi * gj + k;
                if (idxno >= 0 && idxno < A)
                    hit[(long)b * (A + 1) + idxno] = 1;
            }
        }
        bestIouM[m] = bi; bestAM[m] = ba;
    }
    int bf = 0; float bb = bestIouM[0];
    if (bestIouM[1] > bb) { bb = bestIouM[1]; bf = 1; }
    if (bestIouM[2] > bb) { bb = bestIouM[2]; bf = 2; }

    Rec r; r.idx = -1; r.cls = 0;
    r.tx = r.ty = r.tw = r.th = r.wwh = r.tconf = 0.0f;
    if (valid) {
        int idx = ln[bf] + 3 * gim[bf] * gjm[bf] + bestAM[bf];
        if (idx < 0 || idx >= A) idx = -1;             // index A == dropped
        r.idx = idx;
        r.cls = (int)t0;
        r.tx  = t1 * 416.0f; r.ty = t2 * 416.0f;
        r.tw  = t3 * 416.0f; r.th = t4 * 416.0f;
        r.wwh = 2.0f - t3 * t4;
        r.tconf = bb;
    }
    recs[(long)b * N_TGT + n] = r;
}

// ---------------- kernel 2: sparse position losses ----------------------
// acc: [0]=xywh [1]=conf_on [2]=cls [3]=noobj [4]=npos
__global__ void k_pos(const float* __restrict__ pred,
                      const Rec* __restrict__ recs,
                      float* __restrict__ acc,
                      int A) {
    int b = blockIdx.x;
    int n = threadIdx.x;
    if (n >= N_TGT) return;
    const Rec* base = recs + (long)b * N_TGT;
    Rec r = base[n];
    if (r.idx < 0) return;
    // duplicate scatter indices: last write wins -> only largest-n survives
    for (int n2 = n + 1; n2 < N_TGT; ++n2)
        if (base[n2].idx == r.idx) return;
    // tcls is multi-hot across colliding targets: collect class bits
    unsigned cm[3] = {0u, 0u, 0u};
    for (int n2 = 0; n2 < N_TGT; ++n2) {
        Rec q = base[n2];
        if (q.idx == r.idx) {
            int c = q.cls;
            if (c >= 0 && c < NCLS) cm[c >> 5] |= (1u << (c & 31));
        }
    }

    const float* p = pred + ((long)b * A + r.idx) * 85;
    float x = p[0], y = p[1], w = p[2], h = p[3], conf = p[4];

    float w2 = r.wwh * r.wwh;
    float dx = x - r.tx, dy = y - r.ty, dw = w - r.tw, dh = h - r.th;
    float xywh = w2 * (dx*dx + dy*dy + dw*dw + dh*dh);

    float pc = clip_p(conf);
    float conf_on = -(r.tconf * logf(pc) + (1.0f - r.tconf) * logf(1.0f - pc));

    float cls_sum = 0.0f;
    for (int c = 0; c < NCLS; ++c) {
        float pv = clip_p(p[5 + c]);
        bool on = (cm[c >> 5] >> (c & 31)) & 1u;
        cls_sum += on ? -logf(pv) : -logf(1.0f - pv);
    }

    atomicAdd(&acc[0], xywh);
    atomicAdd(&acc[1], conf_on);
    atomicAdd(&acc[2], cls_sum);
    atomicAdd(&acc[4], 1.0f);
}

// ---------------- kernel 3: dense no-object conf BCE --------------------
__global__ void k_noobj(const float* __restrict__ pred,
                        const unsigned char* __restrict__ hit,
                        float* __restrict__ acc,
                        long BA, int A) {
    float s = 0.0f;
    long gs = (long)gridDim.x * blockDim.x;
    for (long e = (long)blockIdx.x * blockDim.x + threadIdx.x; e < BA; e += gs) {
        long pn = e + gs;
        if (pn < BA)
            __builtin_prefetch(&pred[pn * 85 + 4], 0, 1);   // global_prefetch_b8
        float conf = pred[e * 85 + 4];
        long  b    = e / A;                                 // hit idx = e + b
        if (!hit[e + b]) {
            float pv = clip_p(conf);
            s += -logf(1.0f - pv);
        }
        // hit==1 positions: p=clip(0)=EPS -> -log(1-EPS)==0 in fp32
    }
    float wsum = wave_sum_wmma(s);                          // convergent: EXEC all 1s
    if ((threadIdx.x & 31) == 0) atomicAdd(&acc[3], wsum);
}

// ---------------- kernel 4: finalize ------------------------------------
__global__ void k_final(const float* __restrict__ acc,
                        float* __restrict__ out, float BA) {
    float npos = acc[4];
    out[0] = acc[0] / BA                      // loss_x+y+w+h
           + acc[1] / BA                      // conf on-position BCE mean
           + 0.5f * acc[3] / BA               // no-object BCE mean
           + acc[2] / (npos * (float)NCLS);   // class loss
}

extern "C" void kernel_launch(void* const* d_in, const int* in_sizes, int n_in,
                              void* d_out, int out_size, void* d_ws, size_t ws_size,
                              hipStream_t stream) {
    const float* pred    = (const float*)d_in[0];
    const float* tgt     = (const float*)d_in[1];
    const int*   stridep = (const int*)d_in[2];

    int  B  = in_sizes[1] / (N_TGT * 5);                 // 64
    long BA85 = (long)in_sizes[0];
    int  A  = (int)(BA85 / ((long)B * 85));              // 10647
    long BA = (long)B * A;

    // workspace layout
    float*         acc  = (float*)d_ws;                               // 8 f32
    Rec*           recs = (Rec*)((char*)d_ws + 32);                   // B*50*32B
    unsigned char* hit  = (unsigned char*)((char*)d_ws + 32 + (size_t)B * N_TGT * 32);
    long hit_bytes = (long)B * (A + 1);
    long hit_words = (hit_bytes + 3) / 4;

    int ib = (int)((hit_words + 255) / 256);
    if (ib > 1024) ib = 1024;
    k_init<<<ib, 256, 0, stream>>>((unsigned int*)hit, hit_words, acc);

    k_targets<<<B, 64, 0, stream>>>(tgt, stridep, recs, hit, B, A);

    k_pos<<<B, 64, 0, stream>>>(pred, recs, acc, A);

    long want = (BA + 4 * 256 - 1) / (4 * 256);
    int nb = (int)(want > 1024 ? 1024 : (want < 1 ? 1 : want));
    k_noobj<<<nb, 256, 0, stream>>>(pred, hit, acc, BA, A);

    k_final<<<1, 1, 0, stream>>>(acc, (float*)d_out, (float)BA);
}